// MaskLoss_12652973654108
// MI455X (gfx1250) — compile-verified
//
#include <hip/hip_runtime.h>

// MI455X / gfx1250: wave32, WMMA (not MFMA).
// Interp-as-banded-GEMM using V_WMMA_F32_16X16X4_F32 (full fp32 precision).

typedef __attribute__((ext_vector_type(2))) float v2f;  // A/B of 16x16x4 f32: 64 vals / 32 lanes
typedef __attribute__((ext_vector_type(8))) float v8f;  // C/D 16x16 f32: 256 vals / 32 lanes

#define B_   16
#define C_   128
#define H_   128
#define W_   128
#define HW_  (H_ * W_)
#define K_   256
#define S_   1024
#define NOBJ (B_ * K_)      // 4096 objects
#define NGRP (NOBJ / 16)    // 256 row-groups of 16 objects
#define CT_PER_WAVE 16      // 64 column tiles / 4 waves
#define WAVES_PER_BLK 4

__global__ void zero_ws_kernel(float* ws) { ws[0] = 0.0f; }

// weight of channel c for output position with (lo, hi, w) interp params
__device__ __forceinline__ float interp_wt(int c, int lo, int hic, float w) {
    return (c == lo ? 1.0f - w : 0.0f) + (c == hic ? w : 0.0f);
}

__global__ void maskloss_wmma_kernel(const float* __restrict__ shape,
                                     const int*   __restrict__ mask,
                                     const int*   __restrict__ ind,
                                     const float* __restrict__ target,
                                     float*       __restrict__ ws)
{
    const int lane = threadIdx.x & 31;
    const int wave = threadIdx.x >> 5;
    const int g    = blockIdx.x;      // 16-object group; objects o = 16g..16g+15
    const int b    = g >> 4;          // batch index (16 groups per batch since K=256)
    const int col  = lane & 15;       // matrix row (A) / column (B,D) index
    const int hi   = lane >> 4;       // half-wave selector (k_local split)

    // Per-lane gather base: A row m = col, object o = 16g + col.
    const int  o_row = g * 16 + col;
    const float* abase = shape + (long)b * (C_ * HW_) + (long)ind[o_row];

    // Masks for the 8 D rows this lane holds (M = r + 8*hi).
    float m8[8];
#pragma unroll
    for (int r = 0; r < 8; ++r)
        m8[r] = (float)mask[g * 16 + r + 8 * hi];

    float partial = 0.0f;
    const int ct0 = wave * CT_PER_WAVE;

    for (int ci = 0; ci < CT_PER_WAVE; ++ci) {
        const int ct = ct0 + ci;                 // wave-uniform
        const int s  = ct * 16 + col;            // this lane's output position

        // align_corners=False: pos = clip((s+0.5)*C/S - 0.5, 0, C-1), C/S = 1/8
        float pos = fminf(fmaxf(0.125f * (float)s - 0.4375f, 0.0f), 127.0f);
        int   lo  = (int)pos;                    // pos >= 0, trunc == floor
        int   hic = min(lo + 1, 127);
        float w   = pos - (float)lo;

        // Band: tile's 16 columns touch channels [2ct-1, 2ct+2] -> 2 K-tiles.
        const int ktA = max(2 * ct - 1, 0) >> 2;
        const int ktB = min(ktA + 1, 31);
        const float dup = (ktA == 31) ? 0.0f : 1.0f;   // zero duplicate tile at clamp

        const int cA = 4 * ktA + 2 * hi;
        const int cB = 4 * ktB + 2 * hi;

        v2f a1, a2, b1, b2;
        a1.x = abase[(long)cA * HW_];
        a1.y = abase[(long)(cA + 1) * HW_];
        a2.x = abase[(long)cB * HW_];
        a2.y = abase[(long)(cB + 1) * HW_];
        b1.x = interp_wt(cA,     lo, hic, w);
        b1.y = interp_wt(cA + 1, lo, hic, w);
        b2.x = interp_wt(cB,     lo, hic, w) * dup;
        b2.y = interp_wt(cB + 1, lo, hic, w) * dup;

        v8f acc = {};
        // 8 args: (neg_a, A, neg_b, B, c_mod, C, reuse_a, reuse_b)
        acc = __builtin_amdgcn_wmma_f32_16x16x4_f32(false, a1, false, b1,
                                                    (short)0, acc, false, false);
        acc = __builtin_amdgcn_wmma_f32_16x16x4_f32(false, a2, false, b2,
                                                    (short)0, acc, false, false);

        // D element (VGPR r) -> row M = r + 8*hi, col N = col. Compare vs target.
        const float* tcol = target + (long)(g * 16 + 8 * hi) * S_ + s;
#pragma unroll
        for (int r = 0; r < 8; ++r) {
            float t = tcol[(long)r * S_];
            partial += m8[r] * fabsf(acc[r] - t);
        }
    }

    // wave32 butterfly reduction
#pragma unroll
    for (int off = 16; off > 0; off >>= 1)
        partial += __shfl_xor(partial, off, 32);
    if (lane == 0)
        unsafeAtomicAdd(ws, partial);   // global_atomic_add_f32
}

__global__ void finalize_kernel(const int* __restrict__ mask,
                                const float* __restrict__ ws,
                                float* __restrict__ out)
{
    __shared__ int sdata[256];
    const int t = threadIdx.x;
    int s = 0;
    for (int i = t; i < NOBJ; i += 256) s += mask[i];
    sdata[t] = s;
    __syncthreads();
    for (int stride = 128; stride > 0; stride >>= 1) {
        if (t < stride) sdata[t] += sdata[t + stride];
        __syncthreads();
    }
    if (t == 0)
        out[0] = ws[0] / ((float)sdata[0] * (float)S_ + 1e-4f);
}

extern "C" void kernel_launch(void* const* d_in, const int* in_sizes, int n_in,
                              void* d_out, int out_size, void* d_ws, size_t ws_size,
                              hipStream_t stream)
{
    const float* shape  = (const float*)d_in[0];
    // d_in[1] = size (=1024, hardcoded), d_in[2] = saliency (unused by reference)
    const int*   mask   = (const int*)d_in[3];
    const int*   ind    = (const int*)d_in[4];
    const float* target = (const float*)d_in[5];
    float* ws  = (float*)d_ws;
    float* out = (float*)d_out;

    zero_ws_kernel<<<1, 1, 0, stream>>>(ws);
    maskloss_wmma_kernel<<<dim3(NGRP), dim3(32 * WAVES_PER_BLK), 0, stream>>>(
        shape, mask, ind, target, ws);
    finalize_kernel<<<1, 256, 0, stream>>>(mask, ws, out);
}